// MaxCosineSimilarityBlock_7696581394832
// MI455X (gfx1250) — compile-verified
//
#include <hip/hip_runtime.h>

typedef __attribute__((ext_vector_type(2))) float v2f;
typedef __attribute__((ext_vector_type(8))) float v8f;
typedef unsigned int u32x4 __attribute__((ext_vector_type(4)));
typedef int i32x4 __attribute__((ext_vector_type(4)));
typedef int i32x8 __attribute__((ext_vector_type(8)));

#define B_  32
#define C_  8
#define L_  2048
#define K_  512
#define S_  64
#define W_  (L_ - S_ + 1)          // 1985 windows
#define WB_ 16                     // 16 blocks of 128 w-rows cover 2048 >= W
#define KT_ (K_ / 64)              // 8 column tiles (64 k's per block)
#define BSTRIDE 68                 // LDS row stride (floats): 64 + TDM pad (4 DW / 64 DW)

#define LDS_BS_OFF 0                         // byte offset of Bs in group segment
#define LDS_XS_OFF (64 * BSTRIDE * 4)        // byte offset of Xs (17408)

#if defined(__has_builtin)
#if __has_builtin(__builtin_amdgcn_tensor_load_to_lds) && \
    __has_builtin(__builtin_amdgcn_s_wait_tensorcnt)
#define USE_TDM 1
#endif
#endif

// The TDM writes LDS behind the compiler's back (the builtin has no pointer
// operand). Escape the LDS pointer into an asm with a memory clobber so
// GlobalsModRef cannot prove the array is never written and fold the loads.
#define LDS_CLOBBER(p) asm volatile("" :: "v"((void*)(p)) : "memory")

// ---------------- Kernel 0: normalize shapelets into workspace ----------------
__global__ void norm_shapelets_kernel(const float* __restrict__ sh,
                                      float* __restrict__ snn) {
  int r = blockIdx.x * blockDim.x + threadIdx.x;   // row over C*K
  if (r >= C_ * K_) return;
  const float* p = sh + (size_t)r * S_;
  float sum = 0.0f;
#pragma unroll 8
  for (int s = 0; s < S_; ++s) { float v = p[s]; sum += v * v; }
  float inv = 1.0f / fmaxf(sqrtf(sum), 1e-8f);
  float* q = snn + (size_t)r * S_;
#pragma unroll 8
  for (int s = 0; s < S_; ++s) q[s] = p[s] * inv;
}

// ---------------- Kernel 1: per-window inverse L2 norms of x ----------------
__global__ void window_invnorm_kernel(const float* __restrict__ x,
                                      float* __restrict__ invn) {
  int i = blockIdx.x * blockDim.x + threadIdx.x;   // over B*C*W
  if (i >= B_ * C_ * W_) return;
  int w  = i % W_;
  int bc = i / W_;
  const float* p = x + (size_t)bc * L_ + w;
  float sum = 0.0f;
#pragma unroll 8
  for (int s = 0; s < S_; ++s) { float v = p[s]; sum += v * v; }
  invn[i] = 1.0f / fmaxf(sqrtf(sum), 1e-8f);
}

// ---------------- Kernel 2: zero the output (atomic-max identity) ----------------
__global__ void zero_out_kernel(float* __restrict__ out, int n) {
  int i = blockIdx.x * blockDim.x + threadIdx.x;
  if (i < n) out[i] = 0.0f;
}

// ---------------- Kernel 3: TDM-staged fused GEMM + ReLU + max-over-w ----------------
// Block = 128 w-rows (8 waves x 16) x 64 k-cols. Per c: the Tensor Data Mover
// DMAs the 64x64 shapelet tile (with pad_enable producing the stride-68
// conflict-free layout) and the 192-float x segment into LDS; inner loop is
// pure ds_load + v_wmma_f32_16x16x4_f32 with invnrm row-scaling fused in.
__global__ void __launch_bounds__(256) simmax_wmma_kernel(
    const float* __restrict__ x, const float* __restrict__ snn,
    const float* __restrict__ invn, float* __restrict__ out) {
  __shared__ float smem[64 * BSTRIDE + 192];
  float* Bs = smem;                    // 64 k-rows x 64 s, stride 68 (17 KB)
  float* Xs = smem + 64 * BSTRIDE;     // x segment: 128 rows + S window tail

  const int lane = threadIdx.x & 31;
  const int wave = threadIdx.x >> 5;

  const int bid = blockIdx.x;
  const int kt = bid % KT_;
  const int wb = (bid / KT_) % WB_;
  const int b  = bid / (KT_ * WB_);

  const int row  = lane & 15;          // M row (A) / N col (B) held by this lane
  const int kofs = (lane >> 4) * 2;    // lanes 0-15: K={0,1}; lanes 16-31: K={2,3}
  const int wbase = wb * 128;
  const int w  = wbase + wave * 16 + row;
  const int wr = min(w, W_ - 1);       // clamp: duplicate rows are no-ops under max
  const int xoff = wr - wbase;         // local offset into Xs
  const int k0 = kt * 64;

  // Prime the LDS array with a real store so it can never be treated as a
  // never-written global (overwritten by staging before any consuming read).
  if (threadIdx.x == 0) { Bs[0] = 0.0f; Xs[0] = 0.0f; }

  v8f acc0 = {}, acc1 = {}, acc2 = {}, acc3 = {};

  for (int c = 0; c < C_; ++c) {
    LDS_CLOBBER(smem);
    __syncthreads();   // previous tile fully consumed before restaging

#if defined(USE_TDM)
    if (wave == 0) {
      // ---- D# for the 64(k) x 64(s) shapelet tile, padded to stride 68 ----
      const unsigned long long gaB =
          (unsigned long long)(const void*)(snn + ((size_t)c * K_ + k0) * S_);
      u32x4 g0B = { 1u,                                  // count=1, user D#
                    (unsigned)LDS_BS_OFF,                // lds_addr
                    (unsigned)(gaB & 0xffffffffu),       // global_addr[31:0]
                    (unsigned)((gaB >> 32) & 0x01ffffffu) | (2u << 30) }; // [56:32] | type=2
      i32x8 g1B = { (int)((2u << 16)      // data_size = 4B
                        | (1u << 20)      // pad_enable
                        | (5u << 22)      // pad_interval: 64 DWORDs
                        | (3u << 25)),    // pad_amount:   4 DWORDs
                    (int)(64u  << 16),    // tensor_dim0 = 64 (s)
                    (int)(512u << 16),    // tensor_dim0 hi=0 | tensor_dim1 = 512 (k)
                    (int)(64u  << 16),    // tensor_dim1 hi=0 | tile_dim0 = 64
                    (int)64,              // tile_dim1 = 64 | tile_dim2 = 0
                    (int)64,              // tensor_dim0_stride = 64
                    0, 0 };               // dim0_stride hi | dim1_stride = 0
      i32x4 g2 = { 1, 0, 0, 0 };          // tensor_dim2=1; no dim3/iterate
      i32x4 g3 = { 0, 0, 0, 0 };
      i32x8 g4 = { 0, 0, 0, 0, 0, 0, 0, 0 };   // extra group (clang-23 form): zero
      __builtin_amdgcn_tensor_load_to_lds(g0B, g1B, g2, g3, g4, 0);

      // ---- D# for the 192-float x segment (OOB tail reads return zero) ----
      const unsigned long long gaX =
          (unsigned long long)(const void*)(x + ((size_t)b * C_ + c) * L_ + wbase);
      const unsigned remain = (unsigned)(L_ - wbase);    // elements left in the row
      u32x4 g0X = { 1u,
                    (unsigned)LDS_XS_OFF,
                    (unsigned)(gaX & 0xffffffffu),
                    (unsigned)((gaX >> 32) & 0x01ffffffu) | (2u << 30) };
      i32x8 g1X = { (int)(2u << 16),                           // data_size = 4B, no pad
                    (int)((remain & 0xffffu) << 16),           // tensor_dim0 lo
                    (int)(((remain >> 16) & 0xffffu) | (1u << 16)), // dim0 hi | tensor_dim1=1
                    (int)(192u << 16),                         // dim1 hi=0 | tile_dim0=192
                    (int)1,                                    // tile_dim1=1 | tile_dim2=0
                    (int)2048,                                 // tensor_dim0_stride
                    0, 0 };
      __builtin_amdgcn_tensor_load_to_lds(g0X, g1X, g2, g3, g4, 0);

      __builtin_amdgcn_s_wait_tensorcnt(0);
    }
#else
    {
      const int kk = threadIdx.x >> 2;
      const int sb = (threadIdx.x & 3) * 16;
      const float4* src = (const float4*)(snn + ((size_t)c * K_ + k0 + kk) * S_ + sb);
      float4* dst = (float4*)(Bs + kk * BSTRIDE + sb);
      dst[0] = src[0]; dst[1] = src[1]; dst[2] = src[2]; dst[3] = src[3];
      if (threadIdx.x < 192) {
        int j = threadIdx.x;
        Xs[j] = x[((size_t)b * C_ + c) * L_ + min(wbase + j, L_ - 1)];
      }
    }
#endif
    __syncthreads();
    LDS_CLOBBER(smem);   // LDS rewritten by the TDM: force real ds_loads below

    const float inv = invn[((size_t)b * C_ + c) * W_ + wr];
    const float* bsr = Bs + row * BSTRIDE;

#pragma unroll
    for (int s0 = 0; s0 < S_; s0 += 4) {
      const int si = s0 + kofs;
      v2f bf0 = *(const v2f*)(bsr + 0 * 16 * BSTRIDE + si);
      v2f bf1 = *(const v2f*)(bsr + 1 * 16 * BSTRIDE + si);
      v2f bf2 = *(const v2f*)(bsr + 2 * 16 * BSTRIDE + si);
      v2f bf3 = *(const v2f*)(bsr + 3 * 16 * BSTRIDE + si);
      v2f a;
      a.x = Xs[xoff + si] * inv;
      a.y = Xs[xoff + si + 1] * inv;

      acc0 = __builtin_amdgcn_wmma_f32_16x16x4_f32(false, a, false, bf0,
                                                   (short)0, acc0, false, false);
      acc1 = __builtin_amdgcn_wmma_f32_16x16x4_f32(false, a, false, bf1,
                                                   (short)0, acc1, false, false);
      acc2 = __builtin_amdgcn_wmma_f32_16x16x4_f32(false, a, false, bf2,
                                                   (short)0, acc2, false, false);
      acc3 = __builtin_amdgcn_wmma_f32_16x16x4_f32(false, a, false, bf3,
                                                   (short)0, acc3, false, false);
    }
  }

  // Epilogue: scale by 1/C, ReLU (implicit: m starts at 0), max over this wave's
  // 16 rows, then atomic-max across tiles. C/D layout: lanes n and n+16 hold
  // column n; VGPR j = rows {j, j+8}.
  const float scale = 1.0f / (float)C_;
  float* obase = out + (size_t)b * K_ + k0 + row;

  {
    float m = 0.0f;
#pragma unroll
    for (int j = 0; j < 8; ++j) m = fmaxf(m, acc0[j] * scale);
    m = fmaxf(m, __shfl_xor(m, 16, 32));
    if (lane < 16) atomicMax((unsigned int*)(obase + 0 * 16), __float_as_uint(m));
  }
  {
    float m = 0.0f;
#pragma unroll
    for (int j = 0; j < 8; ++j) m = fmaxf(m, acc1[j] * scale);
    m = fmaxf(m, __shfl_xor(m, 16, 32));
    if (lane < 16) atomicMax((unsigned int*)(obase + 1 * 16), __float_as_uint(m));
  }
  {
    float m = 0.0f;
#pragma unroll
    for (int j = 0; j < 8; ++j) m = fmaxf(m, acc2[j] * scale);
    m = fmaxf(m, __shfl_xor(m, 16, 32));
    if (lane < 16) atomicMax((unsigned int*)(obase + 2 * 16), __float_as_uint(m));
  }
  {
    float m = 0.0f;
#pragma unroll
    for (int j = 0; j < 8; ++j) m = fmaxf(m, acc3[j] * scale);
    m = fmaxf(m, __shfl_xor(m, 16, 32));
    if (lane < 16) atomicMax((unsigned int*)(obase + 3 * 16), __float_as_uint(m));
  }
}

extern "C" void kernel_launch(void* const* d_in, const int* in_sizes, int n_in,
                              void* d_out, int out_size, void* d_ws, size_t ws_size,
                              hipStream_t stream) {
  const float* x  = (const float*)d_in[0];   // (B, C, L) f32
  const float* sh = (const float*)d_in[1];   // (C, K, S) f32
  float* out = (float*)d_out;                // (B, 1, K) f32

  // Workspace layout: [normalized shapelets | window inverse norms]
  float* snn  = (float*)d_ws;                        // C*K*S   = 262144 f32 (1 MB)
  float* invn = snn + (size_t)C_ * K_ * S_;          // B*C*W   = 508160 f32 (~2 MB)

  norm_shapelets_kernel<<<(C_ * K_ + 255) / 256, 256, 0, stream>>>(sh, snn);
  window_invnorm_kernel<<<(B_ * C_ * W_ + 255) / 256, 256, 0, stream>>>(x, invn);
  zero_out_kernel<<<(B_ * K_ + 255) / 256, 256, 0, stream>>>(out, B_ * K_);

  // B * WB * KT = 32 * 16 * 8 = 4096 blocks of 256 threads (8 waves).
  simmax_wmma_kernel<<<B_ * WB_ * KT_, 256, 0, stream>>>(x, snn, invn, out);
}